// QuantumLayer_19834158973524
// MI455X (gfx1250) — compile-verified
//
#include <hip/hip_runtime.h>

// CDNA5 / gfx1250, wave32. B = 1,048,576 batches, 4 qubits -> 16-amplitude real state.
// Phase 1 (per thread): mid-state after input RY layer + CNOTs (product state * permutation).
// Phase 2 (per wave):   second RY layer = fixed 16x16 Kronecker matrix U applied via
//                       V_WMMA_F32_16X16X4_F32 chained over K=16, then Z-expectations.

typedef float v2f __attribute__((ext_vector_type(2)));
typedef float v8f __attribute__((ext_vector_type(8)));

#define LDS_ROW_STRIDE 40  // 2*40 % 64 == 16 -> half-wave row groups hit disjoint LDS banks

__global__ __launch_bounds__(256) void quantum_layer_kernel(
    const float* __restrict__ x,      // [B,4]
    const float* __restrict__ qp,     // [4]
    float* __restrict__ out)          // [B,4]
{
  // One 16(K) x 32(N) mid-state tile per wave (8 waves / 256-thread block).
  __shared__ float S[8][16][LDS_ROW_STRIDE];

  const int tid   = threadIdx.x;
  const int lane  = tid & 31;
  const int wave  = tid >> 5;
  const int batch = blockIdx.x * 256 + tid;       // B divisible by 256

  // ---------------- Phase 1: per-thread mid-state ----------------
  const float4 xv = reinterpret_cast<const float4*>(x)[batch];
  float a0[4], a1[4];
  {
    const float xs[4] = {xv.x, xv.y, xv.z, xv.w};
#pragma unroll
    for (int w = 0; w < 4; ++w) {
      // theta/2 = tanh(x) * pi/4 ; per-qubit state after H then RY: ((c-s), (c+s))/sqrt2
      float h = tanhf(xs[w]) * 0.78539816339744830962f;
      float s, c;
      __sincosf(h, &s, &c);
      a0[w] = c - s;
      a1[w] = c + s;
    }
  }

  // mid[m] = pre[sigma01(sigma23(sigma12(m)))] ; pre[k] = 0.25 * prod_w a_{bit_w(k)}[w]
#pragma unroll
  for (int m = 0; m < 16; ++m) {
    int k = m;
    if ((k >> 2) & 1) k ^= 2;   // CNOT(q1 -> q2) (applied last in ref, innermost here)
    if ((k >> 1) & 1) k ^= 1;   // CNOT(q2 -> q3)
    if ((k >> 3) & 1) k ^= 4;   // CNOT(q0 -> q1)
    float amp = 0.25f;
    amp *= ((k >> 3) & 1) ? a1[0] : a0[0];
    amp *= ((k >> 2) & 1) ? a1[1] : a0[1];
    amp *= ((k >> 1) & 1) ? a1[2] : a0[2];
    amp *= ((k >> 0) & 1) ? a1[3] : a0[3];
    S[wave][m][lane] = amp;     // row = mid-state index (K), col = lane (batch within wave)
  }

  // ---------------- Phase 2: U from q_params, WMMA ----------------
  float Cw[4], Sw[4];
#pragma unroll
  for (int w = 0; w < 4; ++w) {
    float th = qp[w] * 0.5f;
    __sincosf(th, &Sw[w], &Cw[w]);
  }

  const int mrow = lane & 15;   // A/B layout: lanes 0-15 and 16-31 both map to M/N = lane%16
  const int hi   = lane >> 4;   // half-wave selects K sub-pair (A & B) and M+8 rows (C/D)

  // A operand: U[m][k], 16x4 f32 chunks; per ISA layout lane holds M=lane%16,
  // VGPR j of chunk q holds K = 4q + 2*hi + j.  U = Kron of RY(theta_w).
  v2f A[4];
#pragma unroll
  for (int q = 0; q < 4; ++q) {
#pragma unroll
    for (int j = 0; j < 2; ++j) {
      const int k = 4 * q + 2 * hi + j;
      float v = 1.0f;
#pragma unroll
      for (int w = 0; w < 4; ++w) {
        const int bm = (mrow >> (3 - w)) & 1;
        const int bk = (k    >> (3 - w)) & 1;
        // RY: [ c  -s ; s  c ]  indexed [bm][bk]
        v *= (bm == bk) ? Cw[w] : (bm ? Sw[w] : -Sw[w]);
      }
      A[q][j] = v;
    }
  }

  const int wbase = blockIdx.x * 256 + wave * 32;

#pragma unroll
  for (int t = 0; t < 2; ++t) {                 // two 16-batch tiles per wave
    // B operand: mid-state columns, 4x16 f32 chunks; lane holds N=lane%16,
    // VGPR j of chunk q holds K = 4q + 2*hi + j (mirrors A layout).
    v2f Bm[4];
#pragma unroll
    for (int q = 0; q < 4; ++q) {
#pragma unroll
      for (int j = 0; j < 2; ++j) {
        const int k = 4 * q + 2 * hi + j;
        Bm[q][j] = S[wave][k][t * 16 + mrow];
      }
    }

    v8f acc = {0.f, 0.f, 0.f, 0.f, 0.f, 0.f, 0.f, 0.f};
#pragma unroll
    for (int q = 0; q < 4; ++q) {
      // D = A*B + C : chain K=16 in four K=4 steps
      acc = __builtin_amdgcn_wmma_f32_16x16x4_f32(
          /*neg_a=*/false, A[q], /*neg_b=*/false, Bm[q],
          /*c_mod=*/(short)0, acc, /*reuse_a=*/false, /*reuse_b=*/false);
    }

    // p(m,n) = amp^2 ; out_i(n) = sum_m z_i(m) p(m,n).
    // Lane row set: m = r + 8*hi  ->  z0 sign = (1-2*hi) (constant per half-wave),
    // z1/z2/z3 depend only on r bits.
    float p[8];
#pragma unroll
    for (int r = 0; r < 8; ++r) p[r] = acc[r] * acc[r];

    float sum = 0.f, z1 = 0.f, z2 = 0.f, z3 = 0.f;
#pragma unroll
    for (int r = 0; r < 8; ++r) {
      sum += p[r];
      z1  += ((r >> 2) & 1) ? -p[r] : p[r];
      z2  += ((r >> 1) & 1) ? -p[r] : p[r];
      z3  += ((r >> 0) & 1) ? -p[r] : p[r];
    }
    float z0 = hi ? -sum : sum;

    // combine row halves (m<8 in lanes 0-15, m>=8 in lanes 16-31, same n)
    z0 += __shfl_xor(z0, 16, 32);
    z1 += __shfl_xor(z1, 16, 32);
    z2 += __shfl_xor(z2, 16, 32);
    z3 += __shfl_xor(z3, 16, 32);

    if (hi == t) {
      const int b = wbase + t * 16 + mrow;
      reinterpret_cast<float4*>(out)[b] = make_float4(z0, z1, z2, z3);
    }
  }
}

extern "C" void kernel_launch(void* const* d_in, const int* in_sizes, int n_in,
                              void* d_out, int out_size, void* d_ws, size_t ws_size,
                              hipStream_t stream) {
  const float* x  = (const float*)d_in[0];   // [B,4] float32
  const float* qp = (const float*)d_in[1];   // [4]   float32
  float* out = (float*)d_out;                // [B,4] float32

  const int nb = in_sizes[0] / 4;            // B = 1,048,576 (divisible by 256)
  const int blocks = nb / 256;
  quantum_layer_kernel<<<blocks, 256, 0, stream>>>(x, qp, out);
}